// IndexerBf16_88235808129227
// MI455X (gfx1250) — compile-verified
//
#include <hip/hip_runtime.h>
#include <hip/hip_bf16.h>

typedef __attribute__((ext_vector_type(2))) float v2f;
typedef __attribute__((ext_vector_type(4))) float v4f;
typedef __attribute__((ext_vector_type(8))) float v8f;

#define B_      64
#define H_      64
#define D_      128
#define NBLK_   128
#define PAGE_   64
#define SEQ_    (NBLK_ * PAGE_)   // 8192
#define QSTR_   132               // padded LDS row stride (dwords): bank = (4m+k)%64, conflict-free

__global__ __launch_bounds__(256)
void mqa_logits_wmma_f32(const float* __restrict__ query,
                         const float* __restrict__ weights,
                         const float* __restrict__ kcache,
                         const int*   __restrict__ block_tables,
                         const int*   __restrict__ seq_lens,
                         float* __restrict__ out)
{
    __shared__ float lds_q[H_ * QSTR_];   // 64 x 132 dwords = 33,792 B
    __shared__ float lds_w[H_];

    const int chunk = blockIdx.x;     // 64 chunks x 128 positions
    const int b     = blockIdx.y;
    const int tid   = threadIdx.x;
    const int wave  = tid >> 5;       // 8 waves (wave32)
    const int lane  = tid & 31;
    const int nlo   = lane & 15;      // N index within 16-wide tile
    const int hi    = lane >> 4;      // half-wave selector

    // ---- stage Q[b] (64x128 f32) into padded LDS + weights (once per block) ----
    {
        const v4f* qsrc = (const v4f*)(query + (size_t)b * H_ * D_);
        #pragma unroll
        for (int j = 0; j < 8; ++j) {
            int f = tid + j * 256;            // 0..2047 float4s
            int m = f >> 5, kv = f & 31;      // row m, 16B-chunk kv
            *(v4f*)&lds_q[m * QSTR_ + kv * 4] = qsrc[f];
        }
        if (tid < H_) lds_w[tid] = weights[b * H_ + tid];
    }
    __syncthreads();

    // ---- this wave's 16 positions (always inside one 64-token page) ----
    const int posBase = chunk * 128 + wave * 16;
    const int page    = block_tables[b * NBLK_ + (posBase >> 6)];
    const int inPage  = posBase & (PAGE_ - 1);
    const int seqLen  = seq_lens[b];

    // ---- preload all B fragments (4x16 KxN per k-step) from HBM ----
    // lane L<16 : vgpr0 = K[n=L][kb+0], vgpr1 = K[n=L][kb+1]
    // lane L>=16: vgpr0 = K[n=L-16][kb+2], vgpr1 = K[n=L-16][kb+3]
    const float* krow = kcache + ((size_t)page * PAGE_ + inPage + nlo) * D_ + hi * 2;
    v2f Bf[32];
    #pragma unroll
    for (int ks = 0; ks < 32; ++ks)
        Bf[ks] = *(const v2f*)(krow + ks * 4);

    // ---- GEMM: D=128 as 32 k-steps, M=64 heads as 4 tiles of v_wmma_f32_16x16x4_f32 ----
    v8f acc0 = {}, acc1 = {}, acc2 = {}, acc3 = {};
    const float* abase = lds_q + nlo * QSTR_ + hi * 2;   // A: lane -> row m, half -> k pair
    #pragma unroll
    for (int ks = 0; ks < 32; ++ks) {
        const int kb = ks * 4;
        v2f a0 = *(const v2f*)(abase + (0 * 16) * QSTR_ + kb);
        v2f a1 = *(const v2f*)(abase + (1 * 16) * QSTR_ + kb);
        v2f a2 = *(const v2f*)(abase + (2 * 16) * QSTR_ + kb);
        v2f a3 = *(const v2f*)(abase + (3 * 16) * QSTR_ + kb);
        acc0 = __builtin_amdgcn_wmma_f32_16x16x4_f32(false, a0, false, Bf[ks], (short)0, acc0, false, false);
        acc1 = __builtin_amdgcn_wmma_f32_16x16x4_f32(false, a1, false, Bf[ks], (short)0, acc1, false, false);
        acc2 = __builtin_amdgcn_wmma_f32_16x16x4_f32(false, a2, false, Bf[ks], (short)0, acc2, false, false);
        acc3 = __builtin_amdgcn_wmma_f32_16x16x4_f32(false, a3, false, Bf[ks], (short)0, acc3, false, false);
    }

    // ---- epilogue: logits[n] = sum_m w[m]*relu(s[m,n]) ----
    // C/D layout: VGPR i holds M=i (lanes 0-15) / M=i+8 (lanes 16-31), N = lane%16
    float part = 0.f;
    #pragma unroll
    for (int i = 0; i < 8; ++i) {
        part = fmaf(fmaxf(acc0[i], 0.f), lds_w[ 0 + hi * 8 + i], part);
        part = fmaf(fmaxf(acc1[i], 0.f), lds_w[16 + hi * 8 + i], part);
        part = fmaf(fmaxf(acc2[i], 0.f), lds_w[32 + hi * 8 + i], part);
        part = fmaf(fmaxf(acc3[i], 0.f), lds_w[48 + hi * 8 + i], part);
    }
    // lane L and L+16 share N, hold complementary head halves -> combine across halves
    float tot = part + __shfl_xor(part, 16, 32);

    const int pos = posBase + nlo;
    if (hi == 0)
        out[(size_t)b * SEQ_ + pos] = (pos < seqLen) ? tot : -__builtin_inff();
}

extern "C" void kernel_launch(void* const* d_in, const int* in_sizes, int n_in,
                              void* d_out, int out_size, void* d_ws, size_t ws_size,
                              hipStream_t stream) {
    const float* query        = (const float*)d_in[0];   // [64,64,128]
    const float* weights      = (const float*)d_in[1];   // [64,64]
    const float* index_k      = (const float*)d_in[2];   // [8192,64,128]
    const int*   block_tables = (const int*)d_in[3];     // [64,128]
    const int*   seq_lens     = (const int*)d_in[4];     // [64]
    float*       out          = (float*)d_out;           // [64,8192]

    dim3 grid(SEQ_ / 128, B_);   // 64 seq-chunks x 64 batches
    dim3 block(256);             // 8 wave32 waves; each wave = 16 positions x 64 heads
    mqa_logits_wmma_f32<<<grid, block, 0, stream>>>(query, weights, index_k,
                                                    block_tables, seq_lens, out);
}